// QMixer_83425444758271
// MI455X (gfx1250) — compile-verified
//
#include <hip/hip_runtime.h>
#include <math.h>

typedef __attribute__((ext_vector_type(2))) float v2f;
typedef __attribute__((ext_vector_type(8))) float v8f;

#define BB   128
#define SS   128
#define AA   8
#define ENN  8
#define EE   16
#define NFAL 32
#define NFEN 16
#define DD   64
#define DA   128
#define MMIX 32
#define HH   64

#define ENT_LD 68   // padded row stride for 16x64 (avoid 64-bank conflicts)
#define QK_LD  132  // padded row stride for 16x128
#define EG_LD  17
#define H1_LD  68
#define XIN_LD 52   // padded row stride for staged 16x48 input matrix

#define WAVES 2

struct WaveLds {
  float ent[EE * ENT_LD];   // entity embeddings 16x64
  float qm [EE * QK_LD];    // q/scale 16x128  (also reused as X staging in phase 0)
  float km [EE * QK_LD];    // k       16x128
  float eg [EE * EG_LD];    // energy / attn 16x16
  float am [EE];            // attn mean per entity
  float aout[DD];           // attention-pooled vector
  float h1 [AA * H1_LD];    // hypernet layer-1 relu out 8x64
  float w1m[AA * MMIX];     // |w1| 8x32
  float wfh[HH];            // wf hidden 64
};

__device__ __forceinline__ v8f wmma_f32(v2f a, v2f b, v8f c) {
  // D = A(16x4) x B(4x16) + C(16x16), fp32
  return __builtin_amdgcn_wmma_f32_16x16x4_f32(false, a, false, b, (short)0, c,
                                               false, false);
}

__global__ __launch_bounds__(WAVES * 32)
void qmix_wmma_kernel(const float* __restrict__ agent_qs,
                      const float* __restrict__ ally_states,
                      const float* __restrict__ enemy_states,
                      const float* __restrict__ W_al, const float* __restrict__ b_al,
                      const float* __restrict__ W_en, const float* __restrict__ b_en,
                      const float* __restrict__ Wq,   const float* __restrict__ bq,
                      const float* __restrict__ Wk,   const float* __restrict__ bk,
                      const float* __restrict__ w1_W1, const float* __restrict__ w1_b1,
                      const float* __restrict__ w1_W2, const float* __restrict__ w1_b2,
                      const float* __restrict__ wf_W1, const float* __restrict__ wf_b1,
                      const float* __restrict__ wf_W2, const float* __restrict__ wf_b2,
                      const float* __restrict__ hb_W,  const float* __restrict__ hb_b,
                      const float* __restrict__ v_W1,  const float* __restrict__ v_b1,
                      const float* __restrict__ v_W2,  const float* __restrict__ v_b2,
                      float* __restrict__ out)
{
  __shared__ WaveLds lds[WAVES];
  const int wave = threadIdx.x >> 5;
  const int lane = threadIdx.x & 31;
  const int tok  = blockIdx.x * WAVES + wave;   // tok = b*S + s, grid covers exactly B*S
  WaveLds& w = lds[wave];

  const int half  = lane >> 4;    // selects K-pair (A/B frags) and row-half (C frags)
  const int mrow  = lane & 15;    // A-frag row index (valid for both lane halves)
  const int ncol  = lane & 15;    // B/C-frag column index
  const int crow0 = half * 8;     // first C row held by this lane

  const v8f ZERO8 = {0.f, 0.f, 0.f, 0.f, 0.f, 0.f, 0.f, 0.f};

  // ---------------- Phase 0: stage X[16x48] into LDS (reuse qm region) ---
  // rows 0..7  : cols 0..31  = ally_states[r, tok, :]   (rest zero)
  // rows 8..15 : cols 32..47 = enemy_states[r-8, tok, :] (rest zero)
  float* xin = w.qm;
  for (int i = lane; i < EE * XIN_LD; i += 32) xin[i] = 0.f;
  __builtin_amdgcn_wave_barrier();
  for (int r = 0; r < AA; ++r)   // 32 lanes load one full ally row, coalesced
    xin[r * XIN_LD + lane] =
        ally_states[((size_t)r * BB * SS + tok) * NFAL + lane];
  {
    const int er = half;         // two enemy rows per iter, all 32 lanes active
    const int ec = lane & 15;
    for (int it = 0; it < 4; ++it) {
      const int r = it * 2 + er;
      xin[(AA + r) * XIN_LD + NFAL + ec] =
          enemy_states[((size_t)r * BB * SS + tok) * NFEN + ec];
    }
  }
  __builtin_amdgcn_wave_barrier();

  // ---------------- Phase 1: entity embeddings (16x48 @ 48x64) ----------
  for (int nt = 0; nt < 4; ++nt) {
    v8f acc = ZERO8;
#pragma unroll
    for (int kt = 0; kt < 8; ++kt) {          // K 0..31 -> W_al rows
      const int kk = kt * 4 + 2 * half;
      v2f a = *(const v2f*)&xin[mrow * XIN_LD + kk];
      v2f b; b.x = W_al[kk * DD + nt * 16 + ncol];
             b.y = W_al[(kk + 1) * DD + nt * 16 + ncol];
      acc = wmma_f32(a, b, acc);
    }
#pragma unroll
    for (int kt = 8; kt < 12; ++kt) {         // K 32..47 -> W_en rows
      const int kk = kt * 4 + 2 * half;
      v2f a = *(const v2f*)&xin[mrow * XIN_LD + kk];
      v2f b; b.x = W_en[(kk - NFAL) * DD + nt * 16 + ncol];
             b.y = W_en[(kk - NFAL + 1) * DD + nt * 16 + ncol];
      acc = wmma_f32(a, b, acc);
    }
    const float bias = (half == 0) ? b_al[nt * 16 + ncol] : b_en[nt * 16 + ncol];
#pragma unroll
    for (int v = 0; v < 8; ++v)
      w.ent[(crow0 + v) * ENT_LD + nt * 16 + ncol] = acc[v] + bias;
  }
  __builtin_amdgcn_wave_barrier();

  // ---------------- Phase 2: Q and K projections (16x128) ----------------
  const float invScale = 0.08838834764831845f;   // 1/sqrt(Da=128)
  for (int nt = 0; nt < 8; ++nt) {
    v8f acc = ZERO8;
#pragma unroll
    for (int kt = 0; kt < 16; ++kt) {
      const int kk = kt * 4 + 2 * half;
      v2f a = *(const v2f*)&w.ent[mrow * ENT_LD + kk];
      v2f b; b.x = Wq[kk * DA + nt * 16 + ncol];
             b.y = Wq[(kk + 1) * DA + nt * 16 + ncol];
      acc = wmma_f32(a, b, acc);
    }
    const float bias = bq[nt * 16 + ncol];
#pragma unroll
    for (int v = 0; v < 8; ++v)
      w.qm[(crow0 + v) * QK_LD + nt * 16 + ncol] = (acc[v] + bias) * invScale;
  }
  for (int nt = 0; nt < 8; ++nt) {
    v8f acc = ZERO8;
#pragma unroll
    for (int kt = 0; kt < 16; ++kt) {
      const int kk = kt * 4 + 2 * half;
      v2f a = *(const v2f*)&w.ent[mrow * ENT_LD + kk];
      v2f b; b.x = Wk[kk * DA + nt * 16 + ncol];
             b.y = Wk[(kk + 1) * DA + nt * 16 + ncol];
      acc = wmma_f32(a, b, acc);
    }
    const float bias = bk[nt * 16 + ncol];
#pragma unroll
    for (int v = 0; v < 8; ++v)
      w.km[(crow0 + v) * QK_LD + nt * 16 + ncol] = acc[v] + bias;
  }
  __builtin_amdgcn_wave_barrier();

  // ---------------- Phase 3: energy = (Q/s) @ K^T (16x16) ----------------
  {
    v8f acc = ZERO8;
#pragma unroll
    for (int kt = 0; kt < 32; ++kt) {
      const int kk = kt * 4 + 2 * half;
      v2f a = *(const v2f*)&w.qm[mrow * QK_LD + kk];
      v2f b = *(const v2f*)&w.km[ncol * QK_LD + kk];   // B[kk][n] = K[n][kk]
      acc = wmma_f32(a, b, acc);
    }
#pragma unroll
    for (int v = 0; v < 8; ++v)
      w.eg[(crow0 + v) * EG_LD + ncol] = acc[v];
  }
  __builtin_amdgcn_wave_barrier();

  // ---------------- Phase 4: softmax over query axis, mean over j --------
  if (lane < 16) {
    const int j = lane;
    float mx = -1e30f;
    for (int i = 0; i < 16; ++i) mx = fmaxf(mx, w.eg[i * EG_LD + j]);
    float z = 0.f;
    for (int i = 0; i < 16; ++i) z += __expf(w.eg[i * EG_LD + j] - mx);
    const float inv = 1.f / z;
    for (int i = 0; i < 16; ++i)
      w.eg[i * EG_LD + j] = __expf(w.eg[i * EG_LD + j] - mx) * inv;
  }
  __builtin_amdgcn_wave_barrier();
  if (lane < 16) {
    const int i = lane;
    float s = 0.f;
    for (int j = 0; j < 16; ++j) s += w.eg[i * EG_LD + j];
    w.am[i] = s * (1.f / 16.f);
  }
  __builtin_amdgcn_wave_barrier();

  // ---------------- Phase 5: attn_out[d] = sum_i am[i]*ent[i][d] ---------
  {
    const int d = lane * 2;
    float s0 = 0.f, s1 = 0.f;
    for (int i = 0; i < 16; ++i) {
      const float a = w.am[i];
      s0 += a * w.ent[i * ENT_LD + d];
      s1 += a * w.ent[i * ENT_LD + d + 1];
    }
    w.aout[d] = s0;
    w.aout[d + 1] = s1;
  }
  __builtin_amdgcn_wave_barrier();

  // ---------------- Phase 6: hypernet w1, layer 1 (8x128 @ 128x64) -------
  // emi[m][k] = k<64 ? attn_out[k] : ally_embed[m][k-64]; rows 8..15 = pad
  const bool  vldA  = (mrow < AA);
  const float maskA = vldA ? 1.f : 0.f;
  const int   mrA   = vldA ? mrow : 0;
  for (int nt = 0; nt < 4; ++nt) {
    v8f acc = ZERO8;
#pragma unroll
    for (int kt = 0; kt < 16; ++kt) {         // attn_out half (K 0..63)
      const int kk = kt * 4 + 2 * half;
      v2f a; a.x = w.aout[kk]; a.y = w.aout[kk + 1];
      v2f b; b.x = w1_W1[kk * HH + nt * 16 + ncol];
             b.y = w1_W1[(kk + 1) * HH + nt * 16 + ncol];
      acc = wmma_f32(a, b, acc);
    }
#pragma unroll
    for (int kt = 0; kt < 16; ++kt) {         // ally-embed half (K 64..127)
      const int kk = kt * 4 + 2 * half;
      v2f t = *(const v2f*)&w.ent[mrA * ENT_LD + kk];
      v2f a; a.x = t.x * maskA; a.y = t.y * maskA;
      v2f b; b.x = w1_W1[(64 + kk) * HH + nt * 16 + ncol];
             b.y = w1_W1[(64 + kk + 1) * HH + nt * 16 + ncol];
      acc = wmma_f32(a, b, acc);
    }
    const float bias = w1_b1[nt * 16 + ncol];
    if (half == 0) {                          // rows 0..7 are the valid agents
#pragma unroll
      for (int v = 0; v < 8; ++v)
        w.h1[v * H1_LD + nt * 16 + ncol] = fmaxf(acc[v] + bias, 0.f);
    }
  }
  __builtin_amdgcn_wave_barrier();

  // ---------------- Phase 7: hypernet w1, layer 2 (8x64 @ 64x32), abs ----
  for (int nt = 0; nt < 2; ++nt) {
    v8f acc = ZERO8;
#pragma unroll
    for (int kt = 0; kt < 16; ++kt) {
      const int kk = kt * 4 + 2 * half;
      v2f t = *(const v2f*)&w.h1[mrA * H1_LD + kk];
      v2f a; a.x = t.x * maskA; a.y = t.y * maskA;
      v2f b; b.x = w1_W2[kk * MMIX + nt * 16 + ncol];
             b.y = w1_W2[(kk + 1) * MMIX + nt * 16 + ncol];
      acc = wmma_f32(a, b, acc);
    }
    const float bias = w1_b2[nt * 16 + ncol];
    if (half == 0) {
#pragma unroll
      for (int v = 0; v < 8; ++v)
        w.w1m[v * MMIX + nt * 16 + ncol] = fabsf(acc[v] + bias);
    }
  }
  __builtin_amdgcn_wave_barrier();

  // ---------------- Phase 8: small heads + final mix ---------------------
  // wf hidden: relu(attn_out @ wf_W1 + wf_b1), 2 outputs per lane
  {
    const int j = lane * 2;
    float s0 = wf_b1[j], s1 = wf_b1[j + 1];
    for (int d = 0; d < DD; ++d) {
      const float a = w.aout[d];
      s0 += a * wf_W1[d * HH + j];
      s1 += a * wf_W1[d * HH + j + 1];
    }
    w.wfh[j]     = fmaxf(s0, 0.f);
    w.wfh[j + 1] = fmaxf(s1, 0.f);
  }
  __builtin_amdgcn_wave_barrier();

  const int m = lane;                          // M = 32 == wave size
  float b1v = hb_b[m];
  for (int d = 0; d < DD; ++d) b1v += w.aout[d] * hb_W[d * MMIX + m];

  float hid = b1v;
  for (int a = 0; a < AA; ++a)
    hid += agent_qs[(size_t)tok * AA + a] * w.w1m[a * MMIX + m];
  hid = (hid > 0.f) ? hid : (__expf(hid) - 1.f);   // ELU

  float wfv = wf_b2[m];
  for (int j = 0; j < HH; ++j) wfv += w.wfh[j] * wf_W2[j * MMIX + m];
  wfv = fabsf(wfv);

  float vh = v_b1[m];
  for (int d = 0; d < DD; ++d) vh += w.aout[d] * v_W1[d * MMIX + m];
  vh = fmaxf(vh, 0.f);

  float part = hid * wfv + vh * v_W2[m];
  for (int off = 16; off > 0; off >>= 1)
    part += __shfl_xor(part, off, 32);

  if (lane == 0) out[tok] = part + v_b2[0];
}

extern "C" void kernel_launch(void* const* d_in, const int* in_sizes, int n_in,
                              void* d_out, int out_size, void* d_ws, size_t ws_size,
                              hipStream_t stream) {
  (void)in_sizes; (void)n_in; (void)out_size; (void)d_ws; (void)ws_size;
  const float* agent_qs     = (const float*)d_in[0];
  const float* ally_states  = (const float*)d_in[1];
  const float* enemy_states = (const float*)d_in[2];
  const float* W_al = (const float*)d_in[3];  const float* b_al = (const float*)d_in[4];
  const float* W_en = (const float*)d_in[5];  const float* b_en = (const float*)d_in[6];
  const float* Wq   = (const float*)d_in[7];  const float* bq   = (const float*)d_in[8];
  const float* Wk   = (const float*)d_in[9];  const float* bk   = (const float*)d_in[10];
  const float* w1_W1 = (const float*)d_in[11]; const float* w1_b1 = (const float*)d_in[12];
  const float* w1_W2 = (const float*)d_in[13]; const float* w1_b2 = (const float*)d_in[14];
  const float* wf_W1 = (const float*)d_in[15]; const float* wf_b1 = (const float*)d_in[16];
  const float* wf_W2 = (const float*)d_in[17]; const float* wf_b2 = (const float*)d_in[18];
  const float* hb_W  = (const float*)d_in[19]; const float* hb_b  = (const float*)d_in[20];
  const float* v_W1  = (const float*)d_in[21]; const float* v_b1  = (const float*)d_in[22];
  const float* v_W2  = (const float*)d_in[23]; const float* v_b2  = (const float*)d_in[24];
  float* out = (float*)d_out;

  const int tokens = BB * SS;               // 16384
  dim3 grid(tokens / WAVES);
  dim3 block(WAVES * 32);
  qmix_wmma_kernel<<<grid, block, 0, stream>>>(
      agent_qs, ally_states, enemy_states,
      W_al, b_al, W_en, b_en, Wq, bq, Wk, bk,
      w1_W1, w1_b1, w1_W2, w1_b2, wf_W1, wf_b1, wf_W2, wf_b2,
      hb_W, hb_b, v_W1, v_b1, v_W2, v_b2, out);
}